// VectorQuantizer_18313740550266
// MI455X (gfx1250) — compile-verified
//
#include <hip/hip_runtime.h>
#include <hip/hip_bf16.h>
#include <math.h>

typedef __attribute__((ext_vector_type(2))) float v2f;
typedef __attribute__((ext_vector_type(8))) float v8f;

#define KCB   1024
#define DDIM  64
#define NROWS 65536
#define LDA   65   // padded LDS row stride (floats) to avoid bank conflicts

// Output layout (concatenated flat, reference return order):
// quantize[16,64,64,64] | vq_loss | perplexity | encodings[N,K] | inds[N,1] | distances[N,K]
static constexpr size_t OFF_Q    = 0;
static constexpr size_t OFF_LOSS = 4194304;
static constexpr size_t OFF_PERP = 4194305;
static constexpr size_t OFF_ENC  = 4194306;                 // 67108864 elems
static constexpr size_t OFF_INDS = OFF_ENC + 67108864;      // 65536 elems
static constexpr size_t OFF_DIST = OFF_INDS + 65536;        // 67108864 elems

__global__ void vq_init(unsigned* __restrict__ counts, float* __restrict__ loss) {
    int t = threadIdx.x;
    if (t < KCB) counts[t] = 0u;
    if (t == 0) *loss = 0.0f;
}

__global__ void vq_en(const float* __restrict__ emb, float* __restrict__ en) {
    int k = blockIdx.x * blockDim.x + threadIdx.x;
    if (k < KCB) {
        const float* e = emb + (size_t)k * DDIM;
        float s = 0.0f;
#pragma unroll
        for (int c = 0; c < DDIM; ++c) s += e[c] * e[c];
        en[k] = s;
    }
}

__global__ __launch_bounds__(256)
void vq_main(const float* __restrict__ latents,   // [16,64,64,64] NCHW
             const float* __restrict__ emb,       // [1024,64]
             const float* __restrict__ en,        // [1024] ||e||^2
             float* __restrict__ out,
             unsigned* __restrict__ counts,
             float* __restrict__ gloss) {
    __shared__ float a_lds[16 * LDA];
    __shared__ float zn[16];
    __shared__ float wmin[8][16];
    __shared__ int   wimin[8][16];
    __shared__ int   fidx[16];
    __shared__ float s_loss;

    const int tid  = threadIdx.x;
    const int wave = tid >> 5;
    const int lane = tid & 31;
    const int row_base = blockIdx.x * 16;

    // ---- stage A tile (16 rows x 64 channels) from NCHW latents into LDS ----
    // n = b*4096 + h*64 + w ; latents addr = b*262144 + c*4096 + (h*64+w)
#pragma unroll
    for (int i = 0; i < 4; ++i) {
        int e = tid + 256 * i;          // 0..1023
        int m = e & 15;                 // row in tile
        int c = e >> 4;                 // channel
        int n = row_base + m;
        int b = n >> 12;
        int hw = n & 4095;
        a_lds[m * LDA + c] = latents[(size_t)b * 262144 + (size_t)c * 4096 + (size_t)hw];
    }
    if (tid == 0) s_loss = 0.0f;
    __syncthreads();

    if (tid < 16) {
        float s = 0.0f;
#pragma unroll
        for (int c = 0; c < DDIM; ++c) { float v = a_lds[tid * LDA + c]; s += v * v; }
        zn[tid] = s;
    }
    __syncthreads();

    // ---- per-lane A fragments for all 16 K-steps (WMMA f32 16x16x4 layout) ----
    // lane<16: K offsets {0,1}; lane>=16: K offsets {2,3}; M = lane%16
    const int mrow = lane & 15;
    const int hi   = lane >> 4;
    v2f afrag[16];
#pragma unroll
    for (int s = 0; s < 16; ++s) {
        int k = s * 4 + hi * 2;
        afrag[s].x = a_lds[mrow * LDA + k];
        afrag[s].y = a_lds[mrow * LDA + k + 1];
    }

    float bmin[8];
    int   bidx[8];
#pragma unroll
    for (int r = 0; r < 8; ++r) { bmin[r] = 3.4e38f; bidx[r] = 0; }

    const float2* e2 = reinterpret_cast<const float2*>(emb);
    float* dist = out + OFF_DIST;
    float* enc  = out + OFF_ENC;

    // ---- 8 column tiles of 16 per wave: full K=1024 covered by the block ----
    for (int t = 0; t < 8; ++t) {
        const int cb   = (wave * 8 + t) * 16;
        const int ncol = cb + mrow;
        v8f acc = {};
#pragma unroll
        for (int s = 0; s < 16; ++s) {
            // B = emb^T: B[k][ncol] = emb[ncol*64 + k]; lane half selects k offset {0,1}/{2,3}
            float2 bv = e2[(size_t)ncol * 32 + (size_t)(s * 2 + hi)];
            v2f b; b.x = bv.x; b.y = bv.y;
            acc = __builtin_amdgcn_wmma_f32_16x16x4_f32(
                false, afrag[s], false, b, (short)0, acc, false, false);
        }
        const float env = en[ncol];
#pragma unroll
        for (int r = 0; r < 8; ++r) {
            const int m = r + hi * 8;                         // C layout: vgpr r, lane half
            const float d = zn[m] + env - 2.0f * acc[r];
            const size_t off = (size_t)(row_base + m) * KCB + (size_t)ncol;
            dist[off] = d;
            enc[off]  = 0.0f;                                 // fused encodings zero-fill
            if (d < bmin[r] || (d == bmin[r] && ncol < bidx[r])) { bmin[r] = d; bidx[r] = ncol; }
        }
    }

    // ---- argmin reduce across the 16 lanes of each half (wave32) ----
#pragma unroll
    for (int r = 0; r < 8; ++r) {
        float v = bmin[r]; int ix = bidx[r];
#pragma unroll
        for (int m = 1; m < 16; m <<= 1) {
            float vo = __shfl_xor(v, m, 32);
            int   io = __shfl_xor(ix, m, 32);
            if (vo < v || (vo == v && io < ix)) { v = vo; ix = io; }
        }
        bmin[r] = v; bidx[r] = ix;
    }
    if ((lane & 15) == 0) {
#pragma unroll
        for (int r = 0; r < 8; ++r) {
            wmin[wave][r + hi * 8]  = bmin[r];
            wimin[wave][r + hi * 8] = bidx[r];
        }
    }
    // order the encodings zero-stores before the one-hot store below
    asm volatile("s_wait_storecnt 0" ::: "memory");
    __syncthreads();

    // ---- final per-row argmin across the 8 waves; write index + one-hot ----
    if (tid < 16) {
        float v = wmin[0][tid]; int ix = wimin[0][tid];
#pragma unroll
        for (int w = 1; w < 8; ++w) {
            float vo = wmin[w][tid]; int io = wimin[w][tid];
            if (vo < v || (vo == v && io < ix)) { v = vo; ix = io; }
        }
        fidx[tid] = ix;
        const int n = row_base + tid;
        out[OFF_INDS + (size_t)n] = (float)ix;
        enc[(size_t)n * KCB + (size_t)ix] = 1.0f;
        atomicAdd(&counts[ix], 1u);
    }
    __syncthreads();

    // ---- quantized gather -> NCHW scatter + MSE partial ----
    {
        const int row  = tid >> 4;      // 16 rows
        const int part = tid & 15;      // 16 threads/row, 4 channels each
        const int ix = fidx[row];
        const int n  = row_base + row;
        const int b  = n >> 12;
        const int hw = n & 4095;
        float ls = 0.0f;
#pragma unroll
        for (int j = 0; j < 4; ++j) {
            const int c = part * 4 + j;
            const float q = emb[(size_t)ix * DDIM + c];
            out[OFF_Q + (size_t)b * 262144 + (size_t)c * 4096 + (size_t)hw] = q;
            const float d = q - a_lds[row * LDA + c];
            ls += d * d;
        }
        atomicAdd(&s_loss, ls);
    }
    __syncthreads();
    if (tid == 0) atomicAdd(gloss, s_loss);
}

__global__ void vq_final(const unsigned* __restrict__ counts,
                         const float* __restrict__ gloss,
                         float* __restrict__ out) {
    __shared__ float red[1024];
    const int t = threadIdx.x;
    const float p = (float)counts[t] * (1.0f / 65536.0f);
    red[t] = -p * logf(p + 1e-10f);
    __syncthreads();
    for (int s = 512; s > 0; s >>= 1) {
        if (t < s) red[t] += red[t + s];
        __syncthreads();
    }
    if (t == 0) {
        out[OFF_PERP] = expf(red[0]);
        out[OFF_LOSS] = 1.25f * (*gloss) * (1.0f / (65536.0f * 64.0f));
    }
}

extern "C" void kernel_launch(void* const* d_in, const int* in_sizes, int n_in,
                              void* d_out, int out_size, void* d_ws, size_t ws_size,
                              hipStream_t stream) {
    const float* latents = (const float*)d_in[0];   // [16,64,64,64] f32
    const float* emb     = (const float*)d_in[1];   // [1024,64] f32
    float* out = (float*)d_out;

    unsigned* counts = (unsigned*)d_ws;                         // 1024 u32
    float*    loss   = (float*)((char*)d_ws + 1024 * 4);        // 1 f32
    float*    en     = (float*)((char*)d_ws + 1025 * 4);        // 1024 f32

    vq_init <<<1, 1024, 0, stream>>>(counts, loss);
    vq_en   <<<4, 256, 0, stream>>>(emb, en);
    vq_main <<<NROWS / 16, 256, 0, stream>>>(latents, emb, en, out, counts, loss);
    vq_final<<<1, 1024, 0, stream>>>(counts, loss, out);
}